// RoIPooling2D_51488067944902
// MI455X (gfx1250) — compile-verified
//
#include <hip/hip_runtime.h>
#include <cstdint>

// RoI adaptive max-pool, P=7, on x[1,512,38,38], rois[128,4] -> out[128,512,7,7] (f32)

#define P        7
#define H        38
#define W        38
#define PLANE    (H * W)      // 1444 floats per channel plane
#define C_TOTAL  512
#define CH       8            // channels staged per workgroup (8*1444*4 = 45 KB LDS)
#define NROI     128
#define SCALE    0.0625f

typedef __attribute__((ext_vector_type(4))) float f4;
typedef __attribute__((ext_vector_type(4))) int   i4;
typedef __attribute__((address_space(1))) i4 gi4;   // global int4
typedef __attribute__((address_space(3))) i4 li4;   // LDS int4

__global__ __launch_bounds__(256)
void roipool7x7_kernel(const float* __restrict__ x,
                       const float* __restrict__ rois,
                       float* __restrict__ out) {
  __shared__ float lds[CH * PLANE];

  const int grp = blockIdx.x;        // channel group 0..63
  const int roi = blockIdx.y;        // 0..127
  const int c0  = grp * CH;
  const int tid = threadIdx.x;

  // ---- Stage CH contiguous channel planes into LDS -------------------------
  // Region is CH*PLANE*4 = 46208 bytes, contiguous, 16B aligned (PLANE*4=5776).
  const float* gsrc = x + (size_t)c0 * PLANE;
  const int nvec = CH * PLANE / 4;   // 2888 float4 transfers

#if __has_builtin(__builtin_amdgcn_global_load_async_to_lds_b128)
  // CDNA5 async path: GLOBAL_LOAD_ASYNC_TO_LDS_B128 (tracked by ASYNCcnt)
  for (int i = tid; i < nvec; i += 256) {
    gi4* gp = (gi4*)(uintptr_t)(gsrc + i * 4);
    li4* lp = (li4*)(uint32_t)(uintptr_t)(&lds[i * 4]);
    __builtin_amdgcn_global_load_async_to_lds_b128(gp, lp, /*offset=*/0, /*cpol=*/0);
  }
#if __has_builtin(__builtin_amdgcn_s_wait_asynccnt)
  __builtin_amdgcn_s_wait_asynccnt(0);
#else
  asm volatile("s_wait_asynccnt 0" ::: "memory");
#endif
#else
  // Fallback: plain vectorized copy (b128 loads + ds_store_b128)
  for (int i = tid; i < nvec; i += 256) {
    ((f4*)lds)[i] = ((const f4*)gsrc)[i];
  }
#endif
  __syncthreads();

  // ---- ROI bin geometry (uniform across block; scalarizes nicely) ---------
  const float* r = rois + roi * 4;
  const int y1 = (int)floorf(r[0] * SCALE);
  const int x1 = (int)floorf(r[1] * SCALE);
  const int y2 = (int)floorf(r[2] * SCALE);
  const int x2 = (int)floorf(r[3] * SCALE);
  const int sh = y2 - y1 + 1;        // >= 1
  const int sw = x2 - x1 + 1;        // >= 1

  // ---- Each thread computes (channel, bin) outputs -------------------------
  for (int task = tid; task < CH * P * P; task += 256) {
    const int c  = task / (P * P);
    const int b  = task - c * (P * P);
    const int ph = b / P;
    const int pw = b - ph * P;

    // start = lo + i*S//P ; end = lo + ceil((i+1)*S/P)   (all args non-negative)
    int hs = y1 + (ph * sh) / P;
    int he = y1 + ((ph + 1) * sh + (P - 1)) / P;
    int ws = x1 + (pw * sw) / P;
    int we = x1 + ((pw + 1) * sw + (P - 1)) / P;
    hs = hs < 0 ? 0 : hs;  he = he > H ? H : he;
    ws = ws < 0 ? 0 : ws;  we = we > W ? W : we;

    float m = -__builtin_huge_valf();
    const float* pl = &lds[c * PLANE];
    for (int h = hs; h < he; ++h) {
      const float* row = pl + h * W;
      for (int w = ws; w < we; ++w) {
        m = fmaxf(m, row[w]);
      }
    }

    out[(((size_t)roi * C_TOTAL + (c0 + c)) * P + ph) * P + pw] = m;
  }
}

extern "C" void kernel_launch(void* const* d_in, const int* in_sizes, int n_in,
                              void* d_out, int out_size, void* d_ws, size_t ws_size,
                              hipStream_t stream) {
  const float* x    = (const float*)d_in[0];   // [1,512,38,38]
  const float* rois = (const float*)d_in[1];   // [128,4]
  float* out        = (float*)d_out;           // [128,512,7,7]

  dim3 grid(C_TOTAL / CH, NROI);               // 64 x 128 = 8192 workgroups
  roipool7x7_kernel<<<grid, 256, 0, stream>>>(x, rois, out);
}